// HungarianMatcher_19782619365959
// MI455X (gfx1250) — compile-verified
//
#include <hip/hip_runtime.h>
#include <cstdint>

// ---------------- problem constants (from setup_inputs) ----------------
constexpr int BS = 16;
constexpr int Q  = 900;
constexpr int NC = 91;           // classes
constexpr int T  = 1600;         // total targets
constexpr int N  = BS * Q;       // 14400 predictions

constexpr float ALPHA = 0.25f;
constexpr float EPS_F = 1e-8f;

// ---------------- tiling ----------------
constexpr int ROWS     = 16;     // predictions per block tile
constexpr int TT       = 256;    // targets per block tile
constexpr int NTHREADS = 256;    // 8 wave32

// LDS byte address of a shared-memory pointer: on gfx1250 the flat shared
// aperture keeps the LDS offset in the low 32 bits (CDNA5 ISA).
__device__ __forceinline__ uint32_t lds_addr_of(const void* p) {
    return (uint32_t)(uintptr_t)p;
}

__device__ __forceinline__ void async_load_lds_b128(uint32_t lds_addr, const void* gaddr) {
    asm volatile("global_load_async_to_lds_b128 %0, %1, off"
                 :: "v"(lds_addr), "v"(gaddr) : "memory");
}
__device__ __forceinline__ void async_load_lds_b32(uint32_t lds_addr, const void* gaddr) {
    asm volatile("global_load_async_to_lds_b32 %0, %1, off"
                 :: "v"(lds_addr), "v"(gaddr) : "memory");
}
__device__ __forceinline__ void wait_asynccnt0() {
    asm volatile("s_wait_asynccnt 0x0" ::: "memory");
}

__global__ __launch_bounds__(NTHREADS)
void hungarian_cost_kernel(const float* __restrict__ pred_logits, // [N, NC]
                           const float* __restrict__ pred_boxes,  // [N, 4] cxcywh
                           const int*   __restrict__ tgt_labels,  // [T]
                           const float* __restrict__ tgt_boxes,   // [T, 4] cxcywh
                           float*       __restrict__ out)         // [N, T]
{
    // ---- LDS tiles ----
    __shared__ float s_costc[ROWS][NC + 1];                 // focal class cost table (+pad)
    __shared__ float s_p[9][ROWS];                          // pred: cx,cy,w,h,x0,y0,x1,y1,area
    __shared__ float s_t[9][TT];                            // tgt:  cx,cy,w,h,x0,y0,x1,y1,area
    __shared__ int   s_tlab[TT];                            // tgt labels
    __shared__ __attribute__((aligned(16))) float s_raw[TT * 4]; // DMA staging (raw cxcywh)

    const int tid = threadIdx.x;
    const int n0  = blockIdx.x * ROWS;   // first prediction row of this tile
    const int t0  = blockIdx.y * TT;     // first target col of this tile

    // ---- Stage 1a: kick off async DMA of the target tile (overlaps VALU below) ----
    {
        int tc = t0 + tid;
        if (tc >= T) tc = T - 1;         // clamp (junk cols are never stored)
        async_load_lds_b128(lds_addr_of(&s_raw[tid * 4]), tgt_boxes + (size_t)tc * 4);
        async_load_lds_b32 (lds_addr_of(&s_tlab[tid]),    tgt_labels + tc);
    }

    // ---- Stage 1b: per-row focal class-cost table: costc[r][c] = pos - neg ----
    for (int i = tid; i < ROWS * NC; i += NTHREADS) {
        const int r = i / NC, c = i % NC;
        const float x = pred_logits[(size_t)(n0 + r) * NC + c];
        const float p = 1.0f / (1.0f + __expf(-x));
        const float pos = ALPHA * (1.0f - p) * (1.0f - p) * (-__logf(p + EPS_F));
        const float neg = (1.0f - ALPHA) * p * p * (-__logf(1.0f - p + EPS_F));
        s_costc[r][c] = pos - neg;
    }

    // ---- Stage 1c: pred boxes -> cxcywh + xyxy + area (SoA) ----
    if (tid < ROWS) {
        const float4 b = reinterpret_cast<const float4*>(pred_boxes)[n0 + tid];
        const float x0 = b.x - 0.5f * b.z, y0 = b.y - 0.5f * b.w;
        const float x1 = b.x + 0.5f * b.z, y1 = b.y + 0.5f * b.w;
        s_p[0][tid] = b.x; s_p[1][tid] = b.y; s_p[2][tid] = b.z; s_p[3][tid] = b.w;
        s_p[4][tid] = x0;  s_p[5][tid] = y0;  s_p[6][tid] = x1;  s_p[7][tid] = y1;
        s_p[8][tid] = (x1 - x0) * (y1 - y0);
    }

    // ---- wait for this wave's DMA, then sync all waves ----
    wait_asynccnt0();
    __syncthreads();

    // ---- Stage 1d: target boxes -> SoA cxcywh + xyxy + area ----
    {
        const float cx = s_raw[tid * 4 + 0], cy = s_raw[tid * 4 + 1];
        const float w  = s_raw[tid * 4 + 2], h  = s_raw[tid * 4 + 3];
        const float x0 = cx - 0.5f * w, y0 = cy - 0.5f * h;
        const float x1 = cx + 0.5f * w, y1 = cy + 0.5f * h;
        s_t[0][tid] = cx; s_t[1][tid] = cy; s_t[2][tid] = w;  s_t[3][tid] = h;
        s_t[4][tid] = x0; s_t[5][tid] = y0; s_t[6][tid] = x1; s_t[7][tid] = y1;
        s_t[8][tid] = (x1 - x0) * (y1 - y0);
    }
    __syncthreads();

    // ---- Stage 2: 4 rows x 4 cols per thread, b128 row stores ----
    const int g     = tid & 63;        // column quad index 0..63
    const int rb    = tid >> 6;        // row block 0..3
    const int tbase = 4 * g;           // col offset inside tile
    const bool valid = (t0 + tbase) < T;   // T and tiles are multiples of 4

    float tcx[4], tcy[4], tw[4], th_[4], tx0[4], ty0[4], tx1[4], ty1[4], tar[4];
    int   lab[4];
#pragma unroll
    for (int j = 0; j < 4; ++j) {
        const int c = tbase + j;
        tcx[j] = s_t[0][c]; tcy[j] = s_t[1][c]; tw[j]  = s_t[2][c]; th_[j] = s_t[3][c];
        tx0[j] = s_t[4][c]; ty0[j] = s_t[5][c]; tx1[j] = s_t[6][c]; ty1[j] = s_t[7][c];
        tar[j] = s_t[8][c]; lab[j] = s_tlab[c];
    }

#pragma unroll
    for (int i = 0; i < 4; ++i) {
        const int r = rb * 4 + i;
        const float pcx = s_p[0][r], pcy = s_p[1][r], pw = s_p[2][r], ph = s_p[3][r];
        const float px0 = s_p[4][r], py0 = s_p[5][r], px1 = s_p[6][r], py1 = s_p[7][r];
        const float par = s_p[8][r];

        float o[4];
#pragma unroll
        for (int j = 0; j < 4; ++j) {
            const float cls = s_costc[r][lab[j]];                       // focal gather
            const float l1  = fabsf(pcx - tcx[j]) + fabsf(pcy - tcy[j]) // L1 in cxcywh
                            + fabsf(pw  - tw[j])  + fabsf(ph  - th_[j]);
            // intersection
            const float ix0 = fmaxf(px0, tx0[j]), iy0 = fmaxf(py0, ty0[j]);
            const float ix1 = fminf(px1, tx1[j]), iy1 = fminf(py1, ty1[j]);
            const float iw = fmaxf(ix1 - ix0, 0.0f), ih = fmaxf(iy1 - iy0, 0.0f);
            const float inter = iw * ih;
            const float uni   = par + tar[j] - inter;
            const float iou   = inter / uni;
            // enclosing box
            const float cx0 = fminf(px0, tx0[j]), cy0 = fminf(py0, ty0[j]);
            const float cx1 = fmaxf(px1, tx1[j]), cy1 = fmaxf(py1, ty1[j]);
            const float cw = fmaxf(cx1 - cx0, 0.0f), ch = fmaxf(cy1 - cy0, 0.0f);
            const float areac = cw * ch;
            const float giou  = iou - (areac - uni) / areac;
            o[j] = l1 + cls - giou;     // W_BBOX*bbox + W_CLASS*class + W_GIOU*(-giou)
        }
        if (valid) {
            float4* dst = reinterpret_cast<float4*>(out + (size_t)(n0 + r) * T + t0 + tbase);
            *dst = float4{o[0], o[1], o[2], o[3]};
        }
    }
}

extern "C" void kernel_launch(void* const* d_in, const int* in_sizes, int n_in,
                              void* d_out, int out_size, void* d_ws, size_t ws_size,
                              hipStream_t stream) {
    (void)in_sizes; (void)n_in; (void)out_size; (void)d_ws; (void)ws_size;
    const float* pred_logits = (const float*)d_in[0];   // [16,900,91] f32
    const float* pred_boxes  = (const float*)d_in[1];   // [16,900,4]  f32
    const int*   tgt_labels  = (const int*)  d_in[2];   // [1600]      i32
    const float* tgt_boxes   = (const float*)d_in[3];   // [1600,4]    f32
    float*       out         = (float*)d_out;           // [16,900,1600] f32

    dim3 grid(N / ROWS, (T + TT - 1) / TT);             // 900 x 7
    hungarian_cost_kernel<<<grid, NTHREADS, 0, stream>>>(
        pred_logits, pred_boxes, tgt_labels, tgt_boxes, out);
}